// BigVGAN_13443247637153
// MI455X (gfx1250) — compile-verified
//
#include <hip/hip_runtime.h>
#include <hip/hip_bf16.h>

// ---------------------------------------------------------------------------
// BigVGAN forward for MI455X (gfx1250, wave32).
// * Channel-mixing convs: implicit GEMM on V_WMMA_F32_16X16X4_F32 (fp32 kept:
//   net is weight-bandwidth bound, ~400MB fp32 weights / 23.3 TB/s ~= 17us).
// * Weight tiles double-buffered in LDS via GLOBAL_LOAD_ASYNC_TO_LDS_B128:
//   next chunk's DMA overlaps current chunk's WMMA loop; ASYNCcnt in-order
//   completion lets us wait with a partial count (s_wait_asynccnt NITER).
// * Kernels templated on K -> compile-time wait immediates + unrolled taps.
// * MAC loop is branch-free (EXEC all-1s, as WMMA requires): clamp + mask.
// ---------------------------------------------------------------------------

typedef float v2f __attribute__((ext_vector_type(2)));
typedef float v8f __attribute__((ext_vector_type(8)));

__device__ __forceinline__ unsigned lds_off(const void* p) {
    // Generic pointer to LDS: addr[31:0] is the LDS byte offset (ISA 10.2).
    return (unsigned)(uintptr_t)p;
}
__device__ __forceinline__ void async_ld_b128(unsigned lds, const float* g) {
    asm volatile("global_load_async_to_lds_b128 %0, %1, off"
                 :: "v"(lds), "v"((unsigned long long)(uintptr_t)g)
                 : "memory");
}
template<int N>
__device__ __forceinline__ void wait_async_le() {
    asm volatile("s_wait_asynccnt %0" :: "i"(N) : "memory");
}

// ---------------------------------------------------------------------------
// Dilated conv1d as WMMA GEMM.  Block = 4 waves; one 16(Cout) x 64(T) slab.
// All 4 waves share the double-buffered LDS weight tile (16 co x 4 ci x K).
// out[co][t] = bias[co] + (res?res[co][t]:0)
//            + sum_ci sum_q w[co][ci][q] * in[ci][t + q*dil - pad]
// Weights (Cout,Cin,K) row-major; activations (C,T) row-major; Cin % 4 == 0.
// Every lane issues exactly one async op per staging iter (dummy-redirected
// when out of range) so the per-wave ASYNCcnt is exactly NITER per chunk.
// ---------------------------------------------------------------------------
template<int K>
__global__ __launch_bounds__(128) void conv1d_wmma(
    const float* __restrict__ in, const float* __restrict__ w,
    const float* __restrict__ bias, const float* __restrict__ res,
    float* __restrict__ out, int Cin, int Cout, int T, int dil, int pad)
{
    constexpr int ROWLEN = 4 * K;                 // floats per co-row per chunk
    constexpr int NFLT   = 16 * ROWLEN;
    constexpr int NITER  = (NFLT + 511) / 512;    // async b128 issues per wave

    __shared__ float sw[2][NFLT];
    __shared__ float dummy[4];

    const int tid  = threadIdx.x;
    const int lane = tid & 31;
    const int wave = tid >> 5;
    const int half = lane >> 4;                   // K-pair select (ISA 7.12.2)
    const int l16  = lane & 15;

    const int co0 = blockIdx.x * 16;
    const int t0  = blockIdx.y * 64 + wave * 16;
    const bool wactive = (t0 < T);                // wave-uniform
    const int tb  = t0 + l16;
    const int ka  = 2 * half;

    // Stage one 16 x 4ci x K weight tile into sw[buf] (coalesced async b128).
    auto stage = [&](int buf, int ci0) {
#pragma unroll
        for (int it = 0; it < NITER; ++it) {
            const int off  = (tid + it * 128) * 4;
            const int offc = min(off, NFLT - 4);
            const int row  = offc / ROWLEN;
            const int wi   = offc - row * ROWLEN;
            const bool ok  = (off < NFLT) & (co0 + row < Cout);
            const float* g = ok ? (w + ((size_t)(co0 + row) * Cin + ci0) * K + wi)
                                : w;                       // safe fallback addr
            const unsigned l = ok ? lds_off(&sw[buf][offc]) : lds_off(&dummy[0]);
            async_ld_b128(l, g);                  // one issue per lane, always
        }
    };

    v8f acc = {};
    const int nchunks = Cin >> 2;

    stage(0, 0);
    for (int c = 0; c < nchunks; ++c) {
        const int buf = c & 1;
        if (c + 1 < nchunks) {
            stage(buf ^ 1, (c + 1) * 4);          // overlap DMA with compute
            wait_async_le<NITER>();               // chunk c resident (in-order)
        } else {
            wait_async_le<0>();
        }
        __syncthreads();

        if (wactive) {
            const int ci0 = c * 4;
            const float* in0 = in + (size_t)(ci0 + ka) * T;
            const float* in1 = in0 + T;
            const float* swr = &sw[buf][l16 * ROWLEN + ka * K];
#pragma unroll
            for (int q = 0; q < K; ++q) {
                v2f a;
                a[0] = swr[q];
                a[1] = swr[K + q];
                const int ti = tb + q * dil - pad;
                const float m = ((unsigned)ti < (unsigned)T) ? 1.0f : 0.0f;
                const int tc = min(max(ti, 0), T - 1);
                v2f b;
                b[0] = m * in0[tc];
                b[1] = m * in1[tc];
                acc = __builtin_amdgcn_wmma_f32_16x16x4_f32(
                    false, a, false, b, (short)0, acc, false, false);
            }
        }
        __syncthreads();                          // protect buf reuse at c+2
    }

    if (wactive) {
        const int t = t0 + l16;
        if (res) {
#pragma unroll
            for (int r = 0; r < 8; ++r) {
                const int co = co0 + r + half * 8;    // C/D: M = r + 8*half
                if (co < Cout && t < T)
                    out[(size_t)co * T + t] = acc[r] + bias[co] + res[(size_t)co * T + t];
            }
        } else {
#pragma unroll
            for (int r = 0; r < 8; ++r) {
                const int co = co0 + r + half * 8;
                if (co < Cout && t < T)
                    out[(size_t)co * T + t] = acc[r] + bias[co];
            }
        }
    }
}

// ---------------------------------------------------------------------------
// ConvTranspose1d as phase-masked WMMA GEMM (stride = 1<<lg).
// out[co][t] = bias[co] + sum_ci sum_q w[ci][co][q] * in[ci][(t+pad-q)>>lg]
//              (taps contribute only when (t+pad-q) & (stride-1) == 0)
// Weights (Cin,Cout,K): per ci, a co-tile is 16*K contiguous floats.
// ---------------------------------------------------------------------------
template<int K>
__global__ __launch_bounds__(128) void convtr1d_wmma(
    const float* __restrict__ in, const float* __restrict__ w,
    const float* __restrict__ bias, float* __restrict__ out,
    int Cin, int Cout, int Tin, int Tout, int lg, long wtotal)
{
    constexpr int CILEN = 16 * K;                 // floats per ci per chunk
    constexpr int NFLT  = 4 * CILEN;
    constexpr int NITER = (NFLT + 511) / 512;

    __shared__ float sw[2][NFLT];
    __shared__ float dummy[4];

    const int tid  = threadIdx.x;
    const int lane = tid & 31;
    const int wave = tid >> 5;
    const int half = lane >> 4;
    const int l16  = lane & 15;

    const int co0 = blockIdx.x * 16;
    const int t0  = blockIdx.y * 64 + wave * 16;
    const bool wactive = (t0 < Tout);
    const int tb  = t0 + l16;

    const int stride = 1 << lg;
    const int pad    = (K - stride) / 2;
    const int ka     = 2 * half;

    auto stage = [&](int buf, int ci0) {
#pragma unroll
        for (int it = 0; it < NITER; ++it) {
            const int off  = (tid + it * 128) * 4;
            const int offc = min(off, NFLT - 4);
            const int ci   = offc / CILEN;
            const int wi   = offc - ci * CILEN;
            const long gidx = ((long)(ci0 + ci) * Cout + co0) * K + wi;
            const bool ok  = (off < NFLT) & (gidx + 4 <= wtotal);
            const float* g = ok ? (w + gidx) : w;
            const unsigned l = ok ? lds_off(&sw[buf][offc]) : lds_off(&dummy[0]);
            async_ld_b128(l, g);
        }
    };

    v8f acc = {};
    const int nchunks = Cin >> 2;

    stage(0, 0);
    for (int c = 0; c < nchunks; ++c) {
        const int buf = c & 1;
        if (c + 1 < nchunks) {
            stage(buf ^ 1, (c + 1) * 4);
            wait_async_le<NITER>();
        } else {
            wait_async_le<0>();
        }
        __syncthreads();

        if (wactive) {
            const int ci0 = c * 4;
            const float* in0 = in + (size_t)(ci0 + ka) * Tin;
            const float* in1 = in0 + Tin;
            const float* swr = &sw[buf][ka * CILEN + l16 * K];
#pragma unroll
            for (int q = 0; q < K; ++q) {
                v2f a;
                a[0] = swr[q];
                a[1] = swr[CILEN + q];
                const int s  = tb + pad - q;
                const int si = s >> lg;
                const bool ok = (s >= 0) & ((s & (stride - 1)) == 0) & (si < Tin);
                const float m = ok ? 1.0f : 0.0f;
                const int sc = min(max(si, 0), Tin - 1);
                v2f b;
                b[0] = m * in0[sc];
                b[1] = m * in1[sc];
                acc = __builtin_amdgcn_wmma_f32_16x16x4_f32(
                    false, a, false, b, (short)0, acc, false, false);
            }
        }
        __syncthreads();
    }

    if (wactive) {
        const int t = t0 + l16;
#pragma unroll
        for (int r = 0; r < 8; ++r) {
            const int co = co0 + r + half * 8;
            if (co < Cout && t < Tout)
                out[(size_t)co * Tout + t] = acc[r] + bias[co];
        }
    }
}

// ---------------------------------------------------------------------------
// Fused AA-up(x2) -> snake_beta -> AA-down(x2).  Same T in/out.
// up[n]   = 2 * sum_{i parity-matched} filt[i] * x[clamp((n+15-i)/2 - 5)]
// s[n]    = up + (1/(e^lb+1e-9)) * sin(up*e^la)^2
// down[t] = sum_{j=0..11} filt[j] * s[clamp(2t+j-5)]
// One block: one channel, 128 output samples; s staged in LDS.
// ---------------------------------------------------------------------------
__global__ __launch_bounds__(128) void aa_snake(
    const float* __restrict__ x, const float* __restrict__ filt,
    const float* __restrict__ log_a, const float* __restrict__ log_b,
    float* __restrict__ out, int C, int T)
{
    __shared__ float s_sh[2 * 128 + 10];
    __shared__ float f_sh[12];

    const int c   = blockIdx.y;
    const int t0  = blockIdx.x * 128;
    const int tid = threadIdx.x;

    if (tid < 12) f_sh[tid] = filt[tid];
    const float a    = __expf(log_a[c]);
    const float binv = 1.0f / (__expf(log_b[c]) + 1e-9f);
    __syncthreads();

    const float* xc = x + (long)c * T;
    const int n0 = 2 * t0 - 5;
    const int ns = 2 * 128 + 10;

    for (int idx = tid; idx < ns; idx += 128) {
        int n = n0 + idx;
        n = min(max(n, 0), 2 * T - 1);           // edge pad of s (down-side)
        const int p  = n + 15;
        const int i0 = p & 1;                    // taps with matching parity
        float up = 0.0f;
#pragma unroll
        for (int r = 0; r < 6; ++r) {
            const int i  = i0 + 2 * r;
            int xi = ((p - i) >> 1) - 5;         // edge pad of x (up-side)
            xi = min(max(xi, 0), T - 1);
            up += f_sh[i] * xc[xi];
        }
        up *= 2.0f;
        const float sn = __sinf(up * a);
        s_sh[idx] = up + binv * sn * sn;
    }
    __syncthreads();

    const int t = t0 + tid;
    if (t < T) {
        float acc = 0.0f;
#pragma unroll
        for (int j = 0; j < 12; ++j) acc += f_sh[j] * s_sh[2 * tid + j];
        out[(long)c * T + t] = acc;
    }
}

// ---------------------------------------------------------------------------
// Small elementwise helpers.
// ---------------------------------------------------------------------------
__global__ void copy_k(float* __restrict__ d, const float* __restrict__ s, int n) {
    int i = blockIdx.x * 256 + threadIdx.x;
    if (i < n) d[i] = s[i];
}
__global__ void add_k(float* __restrict__ d, const float* __restrict__ s, int n) {
    int i = blockIdx.x * 256 + threadIdx.x;
    if (i < n) d[i] += s[i];
}
__global__ void scale_copy_k(float* __restrict__ d, const float* __restrict__ s,
                             float f, int n) {
    int i = blockIdx.x * 256 + threadIdx.x;
    if (i < n) d[i] = s[i] * f;
}

// Final 24->1 conv (k=7, pad=3, no bias) + tanh.
__global__ void post_conv_tanh(const float* __restrict__ in,
                               const float* __restrict__ w,
                               float* __restrict__ out, int T)
{
    int t = blockIdx.x * 256 + threadIdx.x;
    if (t >= T) return;
    float acc = 0.0f;
#pragma unroll
    for (int c = 0; c < 24; ++c)
#pragma unroll
        for (int q = 0; q < 7; ++q) {
            int ti = t + q - 3;
            if (0 <= ti && ti < T) acc += w[c * 7 + q] * in[c * T + ti];
        }
    out[t] = tanhf(acc);
}

// ---------------------------------------------------------------------------
// Host orchestration.
// Flattened input order (insertion-order recursive flatten of setup_inputs):
//   0: mel | 1: prenet_w | 2: prenet_b | 3+2i/4+2i: ups[i].{w,b} (i<6)
//   15 + 8*(i*9 + j*3 + d) + {0..7}: amps[i][j][d].{w1,b1,la1,lb1,w2,b2,la2,lb2}
//   447: post_la | 448: post_lb | 449: post_w | 450: filt
// ---------------------------------------------------------------------------
extern "C" void kernel_launch(void* const* d_in, const int* in_sizes, int n_in,
                              void* d_out, int out_size, void* d_ws, size_t ws_size,
                              hipStream_t stream)
{
    (void)in_sizes; (void)n_in; (void)out_size; (void)ws_size;
    auto fin = [&](int i) { return (const float*)d_in[i]; };

    static const int UPR[6]  = {4, 4, 2, 2, 2, 2};
    static const int UPLG[6] = {2, 2, 1, 1, 1, 1};
    static const int UPK[6]  = {8, 8, 4, 4, 4, 4};
    static const int RBK[3]  = {3, 7, 11};
    static const int DILS[3] = {1, 3, 5};

    const size_t SLOT = 196608;              // max C*T floats
    float* ws = (float*)d_ws;
    float* X  = ws + 0 * SLOT;               // stage activation
    float* XS = ws + 1 * SLOT;               // residual-block sum
    float* Y0 = ws + 2 * SLOT;               // amp-block ping
    float* Y1 = ws + 3 * SLOT;               // amp-block pong
    float* TA = ws + 4 * SLOT;               // fused-act temp
    float* TB = ws + 5 * SLOT;               // conv temp
    float* X2 = ws + 6 * SLOT;               // conv-transpose output

    const float* mel  = fin(0);
    const float* filt = fin(450);

    auto nb = [](long n) { return (unsigned)((n + 255) / 256); };

    auto conv = [&](const float* in, const float* w, const float* b, const float* res,
                    float* out, int Cin, int Cout, int T, int K, int dil, int pad) {
        dim3 g((Cout + 15) / 16, (T + 63) / 64);
        switch (K) {
        case 3:  conv1d_wmma<3> <<<g, 128, 0, stream>>>(in, w, b, res, out, Cin, Cout, T, dil, pad); break;
        case 7:  conv1d_wmma<7> <<<g, 128, 0, stream>>>(in, w, b, res, out, Cin, Cout, T, dil, pad); break;
        case 11: conv1d_wmma<11><<<g, 128, 0, stream>>>(in, w, b, res, out, Cin, Cout, T, dil, pad); break;
        }
    };
    auto convtr = [&](const float* in, const float* w, const float* b, float* out,
                      int Cin, int Cout, int Tin, int Tout, int K, int lg) {
        dim3 g((Cout + 15) / 16, (Tout + 63) / 64);
        const long wt = (long)Cin * Cout * K;
        switch (K) {
        case 4: convtr1d_wmma<4><<<g, 128, 0, stream>>>(in, w, b, out, Cin, Cout, Tin, Tout, lg, wt); break;
        case 8: convtr1d_wmma<8><<<g, 128, 0, stream>>>(in, w, b, out, Cin, Cout, Tin, Tout, lg, wt); break;
        }
    };

    // prenet: 100 -> 1536, k=7, pad=3, T=32
    conv(mel, fin(1), fin(2), nullptr, X, 100, 1536, 32, 7, 1, 3);

    int C = 1536, T = 32;
    for (int i = 0; i < 6; ++i) {
        const int Co = C >> 1, To = T * UPR[i];
        const long n = (long)Co * To;

        convtr(X, fin(3 + 2 * i), fin(4 + 2 * i), X2, C, Co, T, To, UPK[i], UPLG[i]);

        for (int j = 0; j < 3; ++j) {
            const int rk = RBK[j];
            copy_k<<<nb(n), 256, 0, stream>>>(Y0, X2, (int)n);
            float* yc = Y0;
            float* ya = Y1;
            for (int dd = 0; dd < 3; ++dd) {
                const int d    = DILS[dd];
                const int base = 15 + 8 * (i * 9 + j * 3 + dd);
                dim3 gf((To + 127) / 128, Co);

                aa_snake<<<gf, 128, 0, stream>>>(yc, filt, fin(base + 2), fin(base + 3),
                                                 TA, Co, To);
                conv(TA, fin(base + 0), fin(base + 1), nullptr, TB, Co, Co, To,
                     rk, d, (rk * d - d) / 2);
                aa_snake<<<gf, 128, 0, stream>>>(TB, filt, fin(base + 6), fin(base + 7),
                                                 TA, Co, To);
                conv(TA, fin(base + 4), fin(base + 5), yc, ya, Co, Co, To,
                     rk, 1, (rk - 1) / 2);
                float* tmp = yc; yc = ya; ya = tmp;
            }
            if (j == 0) copy_k<<<nb(n), 256, 0, stream>>>(XS, yc, (int)n);
            else        add_k<<<nb(n), 256, 0, stream>>>(XS, yc, (int)n);
        }
        scale_copy_k<<<nb(n), 256, 0, stream>>>(X, XS, 1.0f / 3.0f, (int)n);
        C = Co; T = To;
    }

    // post: aa_snake (C=24, T=8192) -> conv 24->1 (k=7) -> tanh
    {
        dim3 gf((T + 127) / 128, C);
        aa_snake<<<gf, 128, 0, stream>>>(X, filt, fin(447), fin(448), TA, C, T);
        post_conv_tanh<<<nb(T), 256, 0, stream>>>(TA, fin(449), (float*)d_out, T);
    }
}